// WeightedMSELoss_69810398429186
// MI455X (gfx1250) — compile-verified
//
#include <hip/hip_runtime.h>

typedef float v2f __attribute__((ext_vector_type(2)));
typedef float v4f __attribute__((ext_vector_type(4)));
typedef float v8f __attribute__((ext_vector_type(8)));

#define NBLOCK  2048
#define NTHREAD 256
#define NWAVES  (NTHREAD / 32)
#define GRID_N  201
#define ALPHA   0.1f

// Wave32 reduction via V_WMMA_F32_16X16X4_F32:
// A (16x4 f32) = the wave's 64 accumulator values (2 per lane: v0,v1),
// B (4x16 f32) = all ones  ->  D[m][n] = rowsum(A)[m] for every n.
// C/D layout: lane<16 vgpr r holds D[r][lane] (rows 0..7),
//             lane>=16 vgpr r holds D[r+8][lane-16] (rows 8..15).
// So per-lane sum of the 8 D vgprs = half the wave total; xor-shuffle by 16
// combines the halves. Exact fp32 throughout.
__device__ __forceinline__ float wave_reduce_wmma(float a0, float a1) {
    v2f A;  A.x = a0;   A.y = a1;
    v2f Bm; Bm.x = 1.0f; Bm.y = 1.0f;
    v8f C = {};
    C = __builtin_amdgcn_wmma_f32_16x16x4_f32(
        /*neg_a=*/false, A, /*neg_b=*/false, Bm,
        /*c_mod=*/(short)0, C, /*reuse_a=*/false, /*reuse_b=*/false);
    float s = C[0] + C[1] + C[2] + C[3] + C[4] + C[5] + C[6] + C[7];
    s += __shfl_xor(s, 16, 32);
    return s;  // every lane holds the wave total
}

__device__ __forceinline__ float point_term(float p0, float p1, float l0, float l1,
                                            const float* __restrict__ wg) {
    float err = fabsf(p0 - l0) + fabsf(p1 - l1);
    // follow reference op order exactly
    float lat = p0 * 180.0f - 90.0f;
    float lon = p1 * 360.0f - 180.0f;
    int xi = (int)floorf(lat / 0.9f) + 100;
    int yi = (int)floorf(lon / 1.8f) + 100;
    xi = xi < 0 ? 0 : (xi > GRID_N - 1 ? GRID_N - 1 : xi);
    yi = yi < 0 ? 0 : (yi > GRID_N - 1 ? GRID_N - 1 : yi);
    float w = wg[xi * GRID_N + yi];          // RT load: stays hot in L2
    return err * (1.0f + ALPHA - ALPHA * w); // err + err*(a - a*w)
}

__global__ void wmse_partial(const float* __restrict__ pred,
                             const float* __restrict__ lab,
                             const float* __restrict__ wg,
                             float* __restrict__ partial,
                             int n4) {
    const v4f* p4 = (const v4f*)pred;
    const v4f* l4 = (const v4f*)lab;
    float acc0 = 0.0f, acc1 = 0.0f;
    int stride = gridDim.x * blockDim.x;
    for (int i = blockIdx.x * blockDim.x + threadIdx.x; i < n4; i += stride) {
        // NT hints: don't let the 128MB stream evict the 161KB weight grid
        v4f p = __builtin_nontemporal_load(p4 + i);
        v4f l = __builtin_nontemporal_load(l4 + i);
        acc0 += point_term(p.x, p.y, l.x, l.y, wg);
        acc1 += point_term(p.z, p.w, l.z, l.w, wg);
    }

    float wsum = wave_reduce_wmma(acc0, acc1);

    __shared__ float lds[NWAVES];
    int lane = threadIdx.x & 31;
    int wid  = threadIdx.x >> 5;
    if (lane == 0) lds[wid] = wsum;
    __syncthreads();
    if (threadIdx.x == 0) {
        float t = 0.0f;
        for (int i = 0; i < NWAVES; ++i) t += lds[i];
        partial[blockIdx.x] = t;
    }
}

__global__ void wmse_final(const float* __restrict__ partial, int n,
                           float* __restrict__ out, float inv_count) {
    float a = 0.0f;
    for (int i = threadIdx.x; i < n; i += blockDim.x) a += partial[i];

    float wsum = wave_reduce_wmma(a, 0.0f);

    __shared__ float lds[NWAVES];
    int lane = threadIdx.x & 31;
    int wid  = threadIdx.x >> 5;
    if (lane == 0) lds[wid] = wsum;
    __syncthreads();
    if (threadIdx.x == 0) {
        float t = 0.0f;
        for (int i = 0; i < NWAVES; ++i) t += lds[i];
        out[0] = t * inv_count;
    }
}

extern "C" void kernel_launch(void* const* d_in, const int* in_sizes, int n_in,
                              void* d_out, int out_size, void* d_ws, size_t ws_size,
                              hipStream_t stream) {
    const float* pred = (const float*)d_in[0];   // [B,2] fp32
    const float* lab  = (const float*)d_in[1];   // [B,2] fp32
    const float* wg   = (const float*)d_in[2];   // [201,201] fp32

    int total = in_sizes[0];        // B*2 floats
    int n4 = total / 4;             // float4 chunks (2 points each)
    float* partial = (float*)d_ws;  // NBLOCK floats of scratch

    wmse_partial<<<NBLOCK, NTHREAD, 0, stream>>>(pred, lab, wg, partial, n4);
    wmse_final<<<1, NTHREAD, 0, stream>>>(partial, NBLOCK, (float*)d_out,
                                          1.0f / (float)total);
}